// SingelSAGPoolNet_60722247630946
// MI455X (gfx1250) — compile-verified
//
#include <hip/hip_runtime.h>
#include <hip/hip_bf16.h>

// ---------------------------------------------------------------------------
// SAGPool GNN for MI455X (gfx1250, wave32).
// Dense matmuls via V_WMMA_F32_16X16X4_F32 (exact f32, memory-bound anyway).
// Sparse aggregation via per-graph LDS f32 atomics (graph = 1024 nodes fits
// a 128KB LDS accumulator per feature-quarter; whole h matrix is L2-resident).
// ---------------------------------------------------------------------------

#define NB    128          // graphs
#define NPG   1024         // nodes per graph
#define NTOT  (NB * NPG)   // 131072 nodes
#define EPG   4096         // edges per graph (DEG=4)
#define ETOT  (NTOT * 4)   // 524288 edges
#define NHID  128

typedef float v8f __attribute__((ext_vector_type(8)));
typedef float v2f __attribute__((ext_vector_type(2)));

// ---------------------------------------------------------------------------
// GEMM: out[M,128] = A[M,K] @ W[K,128] (+bias)(+relu).  M = grid.x*16.
// Wave32: each wave computes one 16x16 f32 tile with V_WMMA_F32_16X16X4_F32.
// A layout per ISA: lanes0-15 hold M=0..15; VGPR0/1 = K,K+1; lanes16-31 = K+2,K+3.
// B: row striped across lanes within VGPR (same K split as A across half-waves).
// C/D: v8f, VGPR i -> M=i (lanes0-15) / M=8+i (lanes16-31), N=lane&15.
// ---------------------------------------------------------------------------
__global__ __launch_bounds__(256) void k_gemm_wmma(
    const float* __restrict__ A, const float* __restrict__ W,
    const float* __restrict__ bias, float* __restrict__ out,
    int K, int relu)
{
  __shared__ float atile[16 * 256];           // A stripe staged in LDS (K<=256)
  const int m0   = blockIdx.x * 16;
  const int wave = threadIdx.x >> 5;
  const int lane = threadIdx.x & 31;
  const int hi   = lane >> 4;                 // half-wave selects K pair
  const int l15  = lane & 15;
  const int n0   = wave * 16;

  for (int i = threadIdx.x; i < 16 * K; i += 256)
    atile[i] = A[(size_t)(m0 + i / K) * K + (i % K)];
  __syncthreads();

  const float* arow = atile + l15 * K;
  v8f c = {};
  for (int k0 = 0; k0 < K; k0 += 4) {
    v2f a, b;
    a.x = arow[k0 + 2 * hi];
    a.y = arow[k0 + 2 * hi + 1];
    b.x = W[(size_t)(k0 + 2 * hi) * NHID + n0 + l15];
    b.y = W[(size_t)(k0 + 2 * hi + 1) * NHID + n0 + l15];
    c = __builtin_amdgcn_wmma_f32_16x16x4_f32(
        false, a, false, b, (short)0, c, false, false);
  }

  const int col = n0 + l15;
  const float bb = bias ? bias[col] : 0.0f;
#pragma unroll
  for (int i = 0; i < 8; ++i) {
    int row = m0 + i + 8 * hi;
    float v = c[i] + bb;
    if (relu) v = fmaxf(v, 0.0f);
    out[(size_t)row * NHID + col] = v;
  }
}

// ---------------------------------------------------------------------------
// Degree + dinv: deg[d] = sum(mask[s]*mask[d]) + mask[d]; dinv = rsqrt or 0.
// One workgroup per graph; LDS accumulator (ds_add_f32).
// ---------------------------------------------------------------------------
__global__ __launch_bounds__(256) void k_deg(
    const int* __restrict__ src, const int* __restrict__ dst,
    const float* __restrict__ mask, float* __restrict__ dinv)
{
  __shared__ float ldeg[NPG];
  const int g = blockIdx.x, nbase = g * NPG, ebase = g * EPG;
  for (int i = threadIdx.x; i < NPG; i += 256) ldeg[i] = 0.0f;
  __syncthreads();
  for (int i = threadIdx.x; i < EPG; i += 256) {
    int s = src[ebase + i], d = dst[ebase + i];
    float ew = mask[s] * mask[d];
    if (ew != 0.0f) atomicAdd(&ldeg[d - nbase], ew);
  }
  __syncthreads();
  for (int i = threadIdx.x; i < NPG; i += 256) {
    float deg = ldeg[i] + mask[nbase + i];
    dinv[nbase + i] = (deg > 0.0f) ? rsqrtf(deg) : 0.0f;
  }
}

// coef[e] = dinv[s]*dinv[d]*mask[s]*mask[d]
__global__ __launch_bounds__(256) void k_coef(
    const int* __restrict__ src, const int* __restrict__ dst,
    const float* __restrict__ mask, const float* __restrict__ dinv,
    float* __restrict__ coef)
{
  int e = blockIdx.x * 256 + threadIdx.x;
  if (e < ETOT) {
    int s = src[e], d = dst[e];
    coef[e] = dinv[s] * dinv[d] * mask[s] * mask[d];
  }
}

// ---------------------------------------------------------------------------
// Feature aggregation: h[d] = relu( sum_e coef*xw[s] + xw[d]*dinv^2 + b ).
// Block = (graph, feature-quarter); LDS acc 1024x32 f32 = 128KB.
// Each wave handles one edge, its 32 lanes one feature each -> ds_add_f32.
// ---------------------------------------------------------------------------
__global__ __launch_bounds__(256) void k_agg(
    const float* __restrict__ xw, const int* __restrict__ src,
    const int* __restrict__ dst, const float* __restrict__ coef,
    const float* __restrict__ dinv, const float* __restrict__ bias,
    float* __restrict__ hout)
{
  extern __shared__ float acc[];              // NPG * 32 floats
  const int g = blockIdx.x, fq = blockIdx.y;
  const int nbase = g * NPG, ebase = g * EPG, fbase = fq * 32;
  for (int i = threadIdx.x; i < NPG * 32; i += 256) acc[i] = 0.0f;
  __syncthreads();

  const int wave = threadIdx.x >> 5, lane = threadIdx.x & 31;
  for (int e = wave; e < EPG; e += 8) {
    float c = coef[ebase + e];
    if (c != 0.0f) {
      int s    = src[ebase + e];
      int dloc = dst[ebase + e] - nbase;
      atomicAdd(&acc[dloc * 32 + lane],
                xw[(size_t)s * NHID + fbase + lane] * c);
    }
  }
  __syncthreads();

  for (int i = threadIdx.x; i < NPG * 32; i += 256) {
    int n = i >> 5, f = i & 31;
    int node = nbase + n, col = fbase + f;
    float di = dinv[node];
    float v  = acc[i] + xw[(size_t)node * NHID + col] * di * di + bias[col];
    hout[(size_t)node * NHID + col] = fmaxf(v, 0.0f);
  }
}

// xwp[row] = dot(h[row,:], Wp[:,0]) — wave per row, butterfly reduction.
__global__ __launch_bounds__(256) void k_matvec(
    const float* __restrict__ h, const float* __restrict__ Wp,
    float* __restrict__ xwp)
{
  const int wave = threadIdx.x >> 5, lane = threadIdx.x & 31;
  const int row  = blockIdx.x * 8 + wave;
  const float* r = h + (size_t)row * NHID;
  float sum = r[lane] * Wp[lane] + r[lane + 32] * Wp[lane + 32] +
              r[lane + 64] * Wp[lane + 64] + r[lane + 96] * Wp[lane + 96];
  for (int off = 16; off > 0; off >>= 1) sum += __shfl_xor(sum, off, 32);
  if (lane == 0) xwp[row] = sum;
}

// score[d] = sum_e coef*xwp[s] + xwp[d]*dinv^2 + bp  (no relu)
__global__ __launch_bounds__(256) void k_sagg(
    const float* __restrict__ xwp, const int* __restrict__ src,
    const int* __restrict__ dst, const float* __restrict__ coef,
    const float* __restrict__ dinv, const float* __restrict__ bp,
    float* __restrict__ score)
{
  __shared__ float sacc[NPG];
  const int g = blockIdx.x, nbase = g * NPG, ebase = g * EPG;
  for (int i = threadIdx.x; i < NPG; i += 256) sacc[i] = 0.0f;
  __syncthreads();
  for (int i = threadIdx.x; i < EPG; i += 256) {
    float c = coef[ebase + i];
    if (c != 0.0f)
      atomicAdd(&sacc[dst[ebase + i] - nbase], xwp[src[ebase + i]] * c);
  }
  __syncthreads();
  const float b0 = bp[0];
  for (int i = threadIdx.x; i < NPG; i += 256) {
    int node = nbase + i;
    float di = dinv[node];
    score[node] = sacc[i] + xwp[node] * di * di + b0;
  }
}

// ---------------------------------------------------------------------------
// Per-graph top-k: bitonic sort of 1024 (score,index) pairs, descending score,
// ascending index on ties (matches jax.lax.top_k). Emits new mask and the
// fused gate scale[node] = tanh(score)*new_mask.
// ---------------------------------------------------------------------------
__global__ __launch_bounds__(1024) void k_topk(
    const float* __restrict__ score, const float* __restrict__ mask,
    float* __restrict__ newmask, float* __restrict__ scale, int kpool)
{
  __shared__ float ss[NPG];
  __shared__ int   si[NPG];
  __shared__ float nm[NPG];
  const int g = blockIdx.x, t = threadIdx.x, node = g * NPG + t;
  const float sc = score[node];
  ss[t] = (mask[node] > 0.0f) ? sc : -__builtin_inff();
  si[t] = t;
  nm[t] = 0.0f;
  __syncthreads();

  for (int k2 = 2; k2 <= NPG; k2 <<= 1)
    for (int j = k2 >> 1; j > 0; j >>= 1) {
      int ixj = t ^ j;
      if (ixj > t) {
        float sa = ss[t], sb = ss[ixj];
        int ia = si[t], ib = si[ixj];
        bool aFirst = (sa > sb) || (sa == sb && ia < ib);
        bool asc = ((t & k2) == 0);
        if (asc ? !aFirst : aFirst) {
          ss[t] = sb; ss[ixj] = sa; si[t] = ib; si[ixj] = ia;
        }
      }
      __syncthreads();
    }

  if (t < kpool) nm[si[t]] = 1.0f;
  __syncthreads();
  newmask[node] = nm[t];
  scale[node]   = tanhf(sc) * nm[t];
}

// h[i] *= scale[node]
__global__ __launch_bounds__(256) void k_scale(
    float* __restrict__ h, const float* __restrict__ scale)
{
  size_t i = (size_t)blockIdx.x * 256 + threadIdx.x;
  h[i] *= scale[i >> 7];
}

// readout: out_acc[g, 0:128] += colmax over kept nodes; [128:256] += colsum/k
__global__ __launch_bounds__(256) void k_readout(
    const float* __restrict__ h, const float* __restrict__ mask,
    float kdiv, float* __restrict__ oacc)
{
  const int g = blockIdx.x, t = threadIdx.x, nbase = g * NPG;
  if (t < NHID) {
    float mx = -__builtin_inff();
    for (int n = 0; n < NPG; ++n)
      if (mask[nbase + n] > 0.0f)
        mx = fmaxf(mx, h[(size_t)(nbase + n) * NHID + t]);
    oacc[g * 256 + t] += mx;
  } else {
    const int f = t - NHID;
    float sum = 0.0f;
    for (int n = 0; n < NPG; ++n)
      sum += h[(size_t)(nbase + n) * NHID + f];     // h already masked
    oacc[g * 256 + NHID + f] += sum / kdiv;
  }
}

__global__ __launch_bounds__(256) void k_fill(float* p, float v, int n) {
  int i = blockIdx.x * 256 + threadIdx.x;
  if (i < n) p[i] = v;
}

// ---------------------------------------------------------------------------
static void run_layer(const float* in, int K, const float* Wm, const float* b,
                      const float* Wp, const float* bp, int kpool,
                      const int* src, const int* dst,
                      const float* maskIn, float* maskOut,
                      float* h, float* xw, float* dinv, float* coef,
                      float* xwp, float* score, float* scale, float* oacc,
                      hipStream_t stream)
{
  k_deg<<<NB, 256, 0, stream>>>(src, dst, maskIn, dinv);
  k_coef<<<ETOT / 256, 256, 0, stream>>>(src, dst, maskIn, dinv, coef);
  k_gemm_wmma<<<NTOT / 16, 256, 0, stream>>>(in, Wm, nullptr, xw, K, 0);
  dim3 agrid(NB, 4);
  k_agg<<<agrid, 256, NPG * 32 * sizeof(float), stream>>>(
      xw, src, dst, coef, dinv, b, h);
  k_matvec<<<NTOT / 8, 256, 0, stream>>>(h, Wp, xwp);
  k_sagg<<<NB, 256, 0, stream>>>(xwp, src, dst, coef, dinv, bp, score);
  k_topk<<<NB, 1024, 0, stream>>>(score, maskIn, maskOut, scale, kpool);
  k_scale<<<(NTOT * NHID) / 256, 256, 0, stream>>>(h, scale);
  k_readout<<<NB, 256, 0, stream>>>(h, maskOut, (float)kpool, oacc);
}

extern "C" void kernel_launch(void* const* d_in, const int* in_sizes, int n_in,
                              void* d_out, int out_size, void* d_ws, size_t ws_size,
                              hipStream_t stream) {
  const float* x   = (const float*)d_in[0];
  const int*   src = (const int*)d_in[1];
  const int*   dst = src + ETOT;
  // d_in[2] = batch (unused; graphs are contiguous blocks of NPG nodes)
  const float* W1 = (const float*)d_in[3];  const float* b1  = (const float*)d_in[4];
  const float* Wp1= (const float*)d_in[5];  const float* bp1 = (const float*)d_in[6];
  const float* W2 = (const float*)d_in[7];  const float* b2  = (const float*)d_in[8];
  const float* Wp2= (const float*)d_in[9];  const float* bp2 = (const float*)d_in[10];
  const float* W3 = (const float*)d_in[11]; const float* b3  = (const float*)d_in[12];
  const float* Wp3= (const float*)d_in[13]; const float* bp3 = (const float*)d_in[14];
  const float* Wl = (const float*)d_in[15]; const float* bl  = (const float*)d_in[16];

  // Workspace carve-up (floats): 2 big feature buffers + small arrays.
  float* w = (float*)d_ws;
  float* h     = w;                 w += (size_t)NTOT * NHID;
  float* xw    = w;                 w += (size_t)NTOT * NHID;
  float* dinv  = w;                 w += NTOT;
  float* coef  = w;                 w += ETOT;
  float* xwp   = w;                 w += NTOT;
  float* score = w;                 w += NTOT;
  float* maskA = w;                 w += NTOT;
  float* maskB = w;                 w += NTOT;
  float* scale = w;                 w += NTOT;
  float* oacc  = w;                 w += NB * 256;

  k_fill<<<NTOT / 256, 256, 0, stream>>>(maskA, 1.0f, NTOT);
  k_fill<<<(NB * 256 + 255) / 256, 256, 0, stream>>>(oacc, 0.0f, NB * 256);

  run_layer(x, 16,   W1, b1, Wp1, bp1, 512, src, dst, maskA, maskB,
            h, xw, dinv, coef, xwp, score, scale, oacc, stream);
  run_layer(h, NHID, W2, b2, Wp2, bp2, 256, src, dst, maskB, maskA,
            h, xw, dinv, coef, xwp, score, scale, oacc, stream);
  run_layer(h, NHID, W3, b3, Wp3, bp3, 128, src, dst, maskA, maskB,
            h, xw, dinv, coef, xwp, score, scale, oacc, stream);

  // Final: relu(oacc[128,256] @ Wl[256,128] + bl)
  k_gemm_wmma<<<NB / 16, 256, 0, stream>>>(oacc, Wl, bl, (float*)d_out, 256, 1);
}